// SoftMoEGating_74629351735466
// MI455X (gfx1250) — compile-verified
//
#include <hip/hip_runtime.h>
#include <hip/hip_bf16.h>
#include <math.h>

typedef __attribute__((ext_vector_type(2))) float v2f;
typedef __attribute__((ext_vector_type(8))) float v8f;

#define B_   8
#define S_   4096
#define D_   1024
#define KC_  32            // E*P columns
#define BS_  (B_*S_)       // 32768 tokens

// float offsets into d_out (concatenated tuple, return order)
#define O_EW   0u          // expert_weights  [B,S,E]      262144
#define O_EI   262144u     // expert_indices  [B,S,E]      262144
#define O_PW   524288u     // phi_weights     [B,S,E,P]    1048576
#define O_SS   1572864u    // soft_slots      [B,E,P,D]    262144
#define O_EIN  1835008u    // expert_inputs   [B,E,P*D]    262144

// ---------------------------------------------------------------------------
// Trivial outputs: expert_weights = 1/E, expert_indices = e (last axis of E=8)
// ---------------------------------------------------------------------------
__global__ void fill_kernel(float* __restrict__ out) {
  int i = blockIdx.x * blockDim.x + threadIdx.x;   // [0, 262144)
  out[O_EW + i] = 0.125f;
  out[O_EI + i] = (float)(i & 7);
}

// ---------------------------------------------------------------------------
// Pass 1: logits[t, k] = x[t, :] @ phi[:, k] + bias[k]
// One wave per 16-token tile; two 16-wide N tiles share the A fragment.
// A 16x4 fp32: lane<16 -> (M=lane, K=0,1); lane>=16 -> (M=lane-16, K=2,3)
// B 4x16 fp32: lanes index N; K split identically across lane halves/VGPRs.
// ---------------------------------------------------------------------------
__global__ void logits_wmma_kernel(const float* __restrict__ x,
                                   const float* __restrict__ phi,
                                   const float* __restrict__ bias,
                                   float* __restrict__ lg) {
  const int wave = (blockIdx.x * blockDim.x + threadIdx.x) >> 5; // tile id [0,2048)
  const int lane = threadIdx.x & 31;
  const int m    = lane & 15;
  const int hi   = lane >> 4;          // which K half this lane carries
  const int row0 = wave * 16;
  const float* xrow = x + (size_t)(row0 + m) * D_ + 2 * hi;

  v8f acc0 = {}; v8f acc1 = {};
  #pragma unroll 4
  for (int k0 = 0; k0 < D_; k0 += 4) {
    v2f a = *(const v2f*)(xrow + k0);              // 8B load: (K=k0+2hi, k0+2hi+1)
    const int kk = k0 + 2 * hi;
    v2f b0, b1;
    b0.x = phi[(size_t)kk       * KC_ + m];
    b0.y = phi[(size_t)(kk + 1) * KC_ + m];
    b1.x = phi[(size_t)kk       * KC_ + 16 + m];
    b1.y = phi[(size_t)(kk + 1) * KC_ + 16 + m];
    acc0 = __builtin_amdgcn_wmma_f32_16x16x4_f32(false, a, false, b0, (short)0, acc0, false, false);
    acc1 = __builtin_amdgcn_wmma_f32_16x16x4_f32(false, a, false, b1, (short)0, acc1, false, false);
  }

  const float bz0 = bias[m];
  const float bz1 = bias[m + 16];
  #pragma unroll
  for (int r = 0; r < 8; ++r) {
    const int M = r + 8 * hi;                       // C/D layout: vgpr r -> rows r / r+8
    lg[(size_t)(row0 + M) * KC_ + m]      = acc0[r] + bz0;
    lg[(size_t)(row0 + M) * KC_ + 16 + m] = acc1[r] + bz1;
  }
}

// ---------------------------------------------------------------------------
// Pass 2: softmax over the sequence axis, in place in the phi_weights region.
// One block per (batch, group-of-4 columns): float4 loads, 3 L2-resident sweeps.
// ---------------------------------------------------------------------------
__global__ void softmax_kernel(float* __restrict__ lg) {
  const int b   = blockIdx.x >> 3;
  const int g   = blockIdx.x & 7;
  const int tid = threadIdx.x;
  float* base = lg + (size_t)b * S_ * KC_ + g * 4;

  __shared__ float4 red[256];

  // sweep 1: column maxima
  float4 mx = make_float4(-INFINITY, -INFINITY, -INFINITY, -INFINITY);
  for (int s = tid; s < S_; s += 256) {
    float4 v = *(const float4*)(base + (size_t)s * KC_);
    mx.x = fmaxf(mx.x, v.x); mx.y = fmaxf(mx.y, v.y);
    mx.z = fmaxf(mx.z, v.z); mx.w = fmaxf(mx.w, v.w);
  }
  red[tid] = mx; __syncthreads();
  for (int o = 128; o > 0; o >>= 1) {
    if (tid < o) {
      float4 a = red[tid], c = red[tid + o];
      a.x = fmaxf(a.x, c.x); a.y = fmaxf(a.y, c.y);
      a.z = fmaxf(a.z, c.z); a.w = fmaxf(a.w, c.w);
      red[tid] = a;
    }
    __syncthreads();
  }
  const float4 M4 = red[0];
  __syncthreads();

  // sweep 2: sum of exp
  float4 sm = make_float4(0.f, 0.f, 0.f, 0.f);
  for (int s = tid; s < S_; s += 256) {
    float4 v = *(const float4*)(base + (size_t)s * KC_);
    sm.x += __expf(v.x - M4.x); sm.y += __expf(v.y - M4.y);
    sm.z += __expf(v.z - M4.z); sm.w += __expf(v.w - M4.w);
  }
  red[tid] = sm; __syncthreads();
  for (int o = 128; o > 0; o >>= 1) {
    if (tid < o) {
      float4 a = red[tid], c = red[tid + o];
      a.x += c.x; a.y += c.y; a.z += c.z; a.w += c.w;
      red[tid] = a;
    }
    __syncthreads();
  }
  const float4 S4 = red[0];
  __syncthreads();
  const float4 R4 = make_float4(1.f / S4.x, 1.f / S4.y, 1.f / S4.z, 1.f / S4.w);

  // sweep 3: normalize in place
  for (int s = tid; s < S_; s += 256) {
    float4 v = *(const float4*)(base + (size_t)s * KC_);
    v.x = __expf(v.x - M4.x) * R4.x; v.y = __expf(v.y - M4.y) * R4.y;
    v.z = __expf(v.z - M4.z) * R4.z; v.w = __expf(v.w - M4.w) * R4.w;
    *(float4*)(base + (size_t)s * KC_) = v;
  }
}

// ---------------------------------------------------------------------------
// Pass 3: soft_slots[b, k, d] = sum_s pw[b, s, k] * x[b, s, d]
// Per batch: Wt(32 x 4096) * X(4096 x 1024). One wave per 32x16 output strip:
// both 16-row M halves share the B fragment (x), so x is streamed ONCE
// (128 MB, L2-resident after pass 1) instead of twice. 6 b32 loads / 2 wmma
// per K-chunk of 4.
// ---------------------------------------------------------------------------
__global__ void slots_wmma_kernel(const float* __restrict__ x,
                                  const float* __restrict__ pw,
                                  float* __restrict__ out) {
  const int wave = (blockIdx.x * blockDim.x + threadIdx.x) >> 5; // [0,512)
  const int lane = threadIdx.x & 31;
  const int m    = lane & 15;
  const int hi   = lane >> 4;
  const int b    = wave >> 6;            // batch
  const int d0   = (wave & 63) * 16;     // 16-wide D tile
  const size_t tokBase = (size_t)b * S_;

  v8f acc0 = {}; v8f acc1 = {};
  #pragma unroll 4
  for (int s0 = 0; s0 < S_; s0 += 4) {
    const int s = s0 + 2 * hi;
    const size_t r0 = (tokBase + s)     * KC_;
    const size_t r1 = (tokBase + s + 1) * KC_;
    v2f a0, a1, bb;
    a0.x = pw[r0 + m];        a0.y = pw[r1 + m];         // slot cols 0..15
    a1.x = pw[r0 + 16 + m];   a1.y = pw[r1 + 16 + m];    // slot cols 16..31
    bb.x = x[(tokBase + s)     * D_ + d0 + m];           // shared B fragment
    bb.y = x[(tokBase + s + 1) * D_ + d0 + m];
    acc0 = __builtin_amdgcn_wmma_f32_16x16x4_f32(false, a0, false, bb, (short)0, acc0, false, false);
    acc1 = __builtin_amdgcn_wmma_f32_16x16x4_f32(false, a1, false, bb, (short)0, acc1, false, false);
  }

  #pragma unroll
  for (int r = 0; r < 8; ++r) {
    const int M0 = r + 8 * hi;           // rows 0..15 (acc0)
    const int M1 = 16 + M0;              // rows 16..31 (acc1)
    const size_t off0 = ((size_t)b * KC_ + M0) * D_ + d0 + m;
    const size_t off1 = ((size_t)b * KC_ + M1) * D_ + d0 + m;
    out[O_SS  + off0] = acc0[r];         // soft_slots
    out[O_EIN + off0] = acc0[r];         // expert_inputs (same data, reshaped)
    out[O_SS  + off1] = acc1[r];
    out[O_EIN + off1] = acc1[r];
  }
}

extern "C" void kernel_launch(void* const* d_in, const int* in_sizes, int n_in,
                              void* d_out, int out_size, void* d_ws, size_t ws_size,
                              hipStream_t stream) {
  const float* x    = (const float*)d_in[0];   // [8,4096,1024] f32
  const float* phi  = (const float*)d_in[1];   // [1024,32] f32
  const float* bias = (const float*)d_in[2];   // [32] f32
  // d_in[3]=num_experts(8), d_in[4]=num_slots(4): hardcoded above
  float* out = (float*)d_out;

  // trivial outputs
  fill_kernel<<<1024, 256, 0, stream>>>(out);
  // logits -> phi_weights region (used as scratch, then normalized in place)
  logits_wmma_kernel<<<256, 256, 0, stream>>>(x, phi, bias, out + O_PW);
  // softmax over sequence axis, in place
  softmax_kernel<<<64, 256, 0, stream>>>(out + O_PW);
  // weighted token reduction -> soft_slots + expert_inputs (one x sweep)
  slots_wmma_kernel<<<64, 256, 0, stream>>>(x, out + O_PW, out);
}